// VerticalMedian_54262616818099
// MI455X (gfx1250) — compile-verified
//
#include <hip/hip_runtime.h>
#include <hip/hip_bf16.h>

namespace {

constexpr int kW     = 1920;
constexpr int kC     = 3;
constexpr int kN     = 1074;            // output rows (1080 - 6)
constexpr int kRowQ  = (kW * kC) / 4;   // 1440 float4 per image row
constexpr int kRPT   = 8;               // output rows per thread
constexpr int kStage = kRPT + 4;        // input rows staged per block (12)
constexpr int kBlock = 256;

__device__ __forceinline__ float4 vmin4(float4 a, float4 b) {
    return make_float4(fminf(a.x, b.x), fminf(a.y, b.y), fminf(a.z, b.z), fminf(a.w, b.w));
}
__device__ __forceinline__ float4 vmax4(float4 a, float4 b) {
    return make_float4(fmaxf(a.x, b.x), fmaxf(a.y, b.y), fmaxf(a.z, b.z), fmaxf(a.w, b.w));
}

// symmetric reflection into [0, kN): -1->0, -2->1, kN->kN-1, kN+1->kN-2
__device__ __forceinline__ int reflect(int i) {
    i = (i < 0) ? (-1 - i) : i;
    return (i >= kN) ? (2 * kN - 1 - i) : i;
}

// 7-exchange median-of-5 network, 10 min/max (4 one-sided), vectorized over float4.
__device__ __forceinline__ float4 med5(float4 v0, float4 v1, float4 v2, float4 v3, float4 v4) {
    float4 t;
    t  = vmin4(v0, v1); v1 = vmax4(v0, v1); v0 = t;   // sort (v0,v1)
    t  = vmin4(v3, v4); v4 = vmax4(v3, v4); v3 = t;   // sort (v3,v4)
    v3 = vmax4(v0, v3);                               // v0 is min-of-4: discard
    v1 = vmin4(v1, v4);                               // v4 is max-of-4: discard
    t  = vmin4(v1, v2); v2 = vmax4(v1, v2); v1 = t;   // median3(v1,v2,v3)
    v2 = vmin4(v2, v3);
    v2 = vmax4(v1, v2);
    return v2;
}

} // namespace

__global__ __launch_bounds__(kBlock)
void VerticalMedian_54262616818099_kernel(const float* __restrict__ img,
                                          float* __restrict__ out) {
    __shared__ float4 smem[kStage * kBlock];          // 48 KB: 12 staged rows x 256 quads

    const int t  = threadIdx.x;
    const int q  = blockIdx.x * kBlock + t;           // float4 column index
    const int qc = (q < kRowQ) ? q : (kRowQ - 1);     // clamp for in-bounds staging
    const int n0 = blockIdx.y * kRPT;                 // first output row of this block

    // x = img[3:-3] viewed as float4 rows; base 256B-aligned, row = 1440 float4.
    const float4* __restrict__ x = reinterpret_cast<const float4*>(img) + (size_t)3 * kRowQ;
    float4* __restrict__ o = reinterpret_cast<float4*>(out);

    // --- Stage rows n0-2 .. n0+kRPT+1 (reflected) into LDS via the async pipe. ---
    // Each lane fetches its own quad of each row into its own LDS slot (no cross-lane
    // sharing => no barrier needed, only the ASYNCcnt wait).
    const uint32_t lds_base = (uint32_t)(uintptr_t)(&smem[0]);
#pragma unroll
    for (int j = 0; j < kStage; ++j) {
        const int row = reflect(n0 - 2 + j);
        const unsigned long long ga =
            (unsigned long long)(uintptr_t)(x + (size_t)row * kRowQ + qc);
        const uint32_t la = lds_base + (uint32_t)((j * kBlock + t) * (int)sizeof(float4));
        asm volatile("global_load_async_to_lds_b128 %0, %1, off"
                     :: "v"(la), "v"(ga) : "memory");
    }
    asm volatile("s_wait_asynccnt 0x0" ::: "memory");

    if (q >= kRowQ) return;                           // partial last x-tile: no compute

    // --- Rolling 5-row window out of LDS (ds_load_b128), median network, store. ---
    float4 w0 = smem[0 * kBlock + t];
    float4 w1 = smem[1 * kBlock + t];
    float4 w2 = smem[2 * kBlock + t];
    float4 w3 = smem[3 * kBlock + t];

#pragma unroll
    for (int r = 0; r < kRPT; ++r) {
        const int n = n0 + r;
        const float4 w4 = smem[(r + 4) * kBlock + t];
        if (n < kN) {
            o[(size_t)n * kRowQ + q] = med5(w0, w1, w2, w3, w4);
        }
        w0 = w1; w1 = w2; w2 = w3; w3 = w4;           // slide window by one row
    }
}

extern "C" void kernel_launch(void* const* d_in, const int* in_sizes, int n_in,
                              void* d_out, int out_size, void* d_ws, size_t ws_size,
                              hipStream_t stream) {
    (void)in_sizes; (void)n_in; (void)d_ws; (void)ws_size; (void)out_size;
    // d_in[0] = img (1080*1920*3 f32); d_in[1] = mask (unused); d_in[2] = vertical_size (==5, unused)
    const float* img = (const float*)d_in[0];
    float* out = (float*)d_out;

    dim3 block(kBlock, 1, 1);
    dim3 grid((kRowQ + kBlock - 1) / kBlock,          // 6
              (kN + kRPT - 1) / kRPT,                 // 135
              1);
    VerticalMedian_54262616818099_kernel<<<grid, block, 0, stream>>>(img, out);
}